// NPRACHSynch_58205396795567
// MI455X (gfx1250) — compile-verified
//
#include <hip/hip_runtime.h>
#include <hip/hip_bf16.h>
#include <math.h>

// ---------------------------------------------------------------------------
// NPRACH synch on MI455X (gfx1250, wave32)
//   Stage 1: presum(5) + 48-point complex DFT as fp32 WMMA GEMM -> y_sg
//   Stage 2: per-(b,p) hop-product scatter, 1024-pt zero-padded DFT scan,
//            argmax + parabolic interp, frequency-offset estimate
// ---------------------------------------------------------------------------

typedef __attribute__((ext_vector_type(2))) float v2f;
typedef __attribute__((ext_vector_type(8))) float v8f;

#define TWOPI_F 6.28318530717958647692f

#define NB        512       // batch
#define NSG       128       // symbol groups
#define NSC       48        // subcarriers / patterns
#define DFT_N     48
#define SEQ_PER   5
#define SPS       288       // samples per symbol group (48 CP + 5*48)
#define NSAMP     36864     // per-batch samples
#define MROWS     (NB * NSG)   // 65536 GEMM rows
#define FFTN      1024
#define VLEN      13
#define NPROB     (NB * NSC)   // 24576 detection problems

// ---------------------------------------------------------------------------
// Kernel 1: 16-row tile, 3 waves (one 16-col N-tile each), K=48 in steps of 4.
// Complex GEMM with W[n,k] = exp(-2*pi*i*n*k/48) = cos - i*sin:
//   Cr = Ar*cos + Ai*sin ;  Ci = Ai*cos - Ar*sin
// ---------------------------------------------------------------------------
__global__ __launch_bounds__(96) void nprach_dft48_wmma(
    const float2* __restrict__ y, float* __restrict__ ysr, float* __restrict__ ysi) {

  __shared__ float Ar[16][50];    // presummed rows, real
  __shared__ float Ai[16][50];    // presummed rows, imag
  __shared__ float Wc[48][49];    // cos(2*pi*n*k/48), [n][k]
  __shared__ float Wsn[48][49];   // sin(2*pi*n*k/48), [n][k]

  const int tid = threadIdx.x;

  // Twiddle tables
  for (int idx = tid; idx < 48 * 48; idx += 96) {
    int n = idx / 48, k = idx % 48;
    float s, c;
    __sincosf((TWOPI_F / 48.0f) * (float)(n * k), &s, &c);
    Wc[n][k]  = c;
    Wsn[n][k] = s;
  }

  const int m0 = blockIdx.x * 16;  // first GEMM row of this tile

  // Presum the 5 sequences of each row into LDS (split re/im)
  for (int e = tid; e < 16 * 48; e += 96) {
    int r = e / 48, n = e % 48;
    int m = m0 + r;
    int b = m >> 7, sg = m & (NSG - 1);
    const float2* src = y + (size_t)b * NSAMP + sg * SPS + DFT_N + n;
    float sr = 0.f, si = 0.f;
#pragma unroll
    for (int w = 0; w < SEQ_PER; ++w) {
      float2 v = src[w * DFT_N];
      sr += v.x;
      si += v.y;
    }
    Ar[r][n] = sr;
    Ai[r][n] = si;
  }
  // Streaming hint for the next tile (global_prefetch_b8)
  if (tid < 16) {
    int m = m0 + 16 + tid;
    if (m < MROWS) {
      int b = m >> 7, sg = m & (NSG - 1);
      __builtin_prefetch((const void*)(y + (size_t)b * NSAMP + sg * SPS + DFT_N), 0, 1);
    }
  }
  __syncthreads();

  const int wave = tid >> 5;
  const int lane = tid & 31;
  const int hlf  = lane >> 4;   // half-wave select
  const int lm   = lane & 15;
  const int n0   = wave * 16;   // output-frequency tile

  v8f cr = {};
  v8f ci = {};
#pragma unroll
  for (int k0 = 0; k0 < 48; k0 += 4) {
    // A 16x4 fp32 fragment: VGPR0 -> K = k0 + 2*hlf, VGPR1 -> K = k0 + 2*hlf + 1
    v2f a_r, a_i, bwc, bws;
    a_r.x = Ar[lm][k0 + 2 * hlf];
    a_r.y = Ar[lm][k0 + 2 * hlf + 1];
    a_i.x = Ai[lm][k0 + 2 * hlf];
    a_i.y = Ai[lm][k0 + 2 * hlf + 1];
    v2f nar = -a_r;
    // B 4x16 fp32 fragment, mirrored K striping
    bwc.x = Wc[k0 + 2 * hlf][n0 + lm];
    bwc.y = Wc[k0 + 2 * hlf + 1][n0 + lm];
    bws.x = Wsn[k0 + 2 * hlf][n0 + lm];
    bws.y = Wsn[k0 + 2 * hlf + 1][n0 + lm];

    cr = __builtin_amdgcn_wmma_f32_16x16x4_f32(false, a_r, false, bwc, (short)0, cr, false, false);
    cr = __builtin_amdgcn_wmma_f32_16x16x4_f32(false, a_i, false, bws, (short)0, cr, false, false);
    ci = __builtin_amdgcn_wmma_f32_16x16x4_f32(false, a_i, false, bwc, (short)0, ci, false, false);
    ci = __builtin_amdgcn_wmma_f32_16x16x4_f32(false, nar, false, bws, (short)0, ci, false, false);
  }

  // C/D layout: VGPR j, lanes 0-15 -> M=j, lanes 16-31 -> M=j+8 ; N = lm
  const float inv48 = 1.0f / 48.0f;
#pragma unroll
  for (int j = 0; j < 8; ++j) {
    size_t row = (size_t)(m0 + j + 8 * hlf);
    int    col = n0 + lm;
    ysr[row * 48 + col] = cr[j] * inv48;
    ysi[row * 48 + col] = ci[j] * inv48;
  }
}

// ---------------------------------------------------------------------------
// Kernel 2: one wave per (b,p). 8 waves / block.
// ---------------------------------------------------------------------------
__global__ __launch_bounds__(256) void nprach_detect(
    const float* __restrict__ ysr, const float* __restrict__ ysi,
    const int* __restrict__ pats, const int* __restrict__ steps,
    const float* __restrict__ tau, float* __restrict__ out) {

  __shared__ float acc[8][40];  // per wave: [2h],[2h+1]=v re/im, [26+h]=count

  const int tid  = threadIdx.x;
  const int wave = tid >> 5;
  const int lane = tid & 31;
  const int pid  = blockIdx.x * 8 + wave;  // 0..NPROB-1
  const int b    = pid / NSC;
  const int p    = pid % NSC;
  float* w = acc[wave];

  for (int t = lane; t < 39; t += 32) w[t] = 0.f;

  // Phase 1: z[i] = y_pr[i] * conj(y_pr[i+1]) scattered into bin h=step+6
#pragma unroll
  for (int t = 0; t < 4; ++t) {
    int i = lane + t * 32;
    if (i < NSG - 1) {
      int c0 = pats[p * NSG + i];
      int c1 = pats[p * NSG + i + 1];
      size_t r0 = (size_t)(b * NSG + i) * 48;
      size_t r1 = r0 + 48;
      float ar = ysr[r0 + c0], ai = ysi[r0 + c0];
      float br = ysr[r1 + c1], bi = ysi[r1 + c1];
      float zr = ar * br + ai * bi;   // a * conj(b)
      float zi = ai * br - ar * bi;
      int h = steps[p * (NSG - 1) + i] + 6;
      atomicAdd(&w[2 * h], zr);       // ds_add_f32
      atomicAdd(&w[2 * h + 1], zi);
      atomicAdd(&w[26 + h], 1.0f);
    }
  }
  __syncthreads();

  float vr[VLEN], vi[VLEN], cn[VLEN];
#pragma unroll
  for (int h = 0; h < VLEN; ++h) {
    vr[h] = w[2 * h];
    vi[h] = w[2 * h + 1];
    cn[h] = w[26 + h];
  }

  // Phase 2: 1024-pt DFT magnitude scan, 32 k's per lane, twiddle recurrence
  float best = -1.0f;
  int bk = 0;
  for (int t = 0; t < 32; ++t) {
    int k = lane + t * 32;
    float sb, cb;
    __sincosf(-(TWOPI_F / (float)FFTN) * (float)k, &sb, &cb);
    float tr = 1.f, ti = 0.f, Vr = 0.f, Vi = 0.f;
#pragma unroll
    for (int h = 0; h < VLEN; ++h) {
      Vr += vr[h] * tr - vi[h] * ti;
      Vi += vr[h] * ti + vi[h] * tr;
      float ntr = tr * cb - ti * sb;
      ti = tr * sb + ti * cb;
      tr = ntr;
    }
    float m2 = Vr * Vr + Vi * Vi;
    if (m2 > best) { best = m2; bk = k; }  // ascending k: keeps first max
  }
  // wave argmax reduction (prefer smaller k on ties, matching jnp.argmax)
  for (int off = 16; off > 0; off >>= 1) {
    float ob = __shfl_xor(best, off, 32);
    int   ok = __shfl_xor(bk, off, 32);
    if (ob > best || (ob == best && ok < bk)) { best = ob; bk = ok; }
  }

  // Phase 3: |V| at k-1, k, k+1 (mod 1024), parabolic interpolation
  auto mag = [&](int k) -> float {
    float sb, cb;
    __sincosf(-(TWOPI_F / (float)FFTN) * (float)k, &sb, &cb);
    float tr = 1.f, ti = 0.f, Vr = 0.f, Vi = 0.f;
#pragma unroll
    for (int h = 0; h < VLEN; ++h) {
      Vr += vr[h] * tr - vi[h] * ti;
      Vi += vr[h] * ti + vi[h] * tr;
      float ntr = tr * cb - ti * sb;
      ti = tr * sb + ti * cb;
      tr = ntr;
    }
    return sqrtf(Vr * Vr + Vi * Vi) * (1.0f / (float)FFTN);
  };
  float xc  = mag(bk);
  float xm  = mag((bk + FFTN - 1) & (FFTN - 1));
  float xp  = mag((bk + 1) & (FFTN - 1));
  float eps = 0.5f * (xp - xm) / (2.0f * xc - xm - xp);
  float toa = ((float)bk + eps) / ((float)FFTN * 3750.0f);

  // Phase 4: f_off = angle( sum_h cnt[h]*e^{i*c*(h-6)} * conj(v[h]) ) / (2*pi*288)
  float c = TWOPI_F * toa * 3750.0f;  // 2*pi*toa*BW/48, BW = 48*3750
  float Sr = 0.f, Si = 0.f;
#pragma unroll
  for (int h = 0; h < VLEN; ++h) {
    float s, co;
    __sincosf(c * (float)(h - 6), &s, &co);
    Sr += cn[h] * (co * vr[h] + s * vi[h]);
    Si += cn[h] * (s * vr[h] - co * vi[h]);
  }
  float foff = atan2f(Si, Sr) / (TWOPI_F * (float)SPS);

  if (lane == 0) {
    int o = b * NSC + p;
    out[o]             = (xc > tau[p]) ? 1.0f : 0.0f;
    out[NPROB + o]     = toa;
    out[2 * NPROB + o] = foff;
  }
}

// ---------------------------------------------------------------------------
extern "C" void kernel_launch(void* const* d_in, const int* in_sizes, int n_in,
                              void* d_out, int out_size, void* d_ws, size_t ws_size,
                              hipStream_t stream) {
  (void)in_sizes; (void)n_in; (void)out_size; (void)ws_size;
  const float2* y     = (const float2*)d_in[0];
  // d_in[1] seq_indices is analytic (sg*288 + 48 + w*48 + n) -> unused
  const int*    pats  = (const int*)d_in[2];
  const int*    steps = (const int*)d_in[3];
  const float*  tau   = (const float*)d_in[4];

  float* ysr = (float*)d_ws;
  float* ysi = ysr + (size_t)MROWS * 48;

  nprach_dft48_wmma<<<MROWS / 16, 96, 0, stream>>>(y, ysr, ysi);
  nprach_detect<<<NPROB / 8, 256, 0, stream>>>(ysr, ysi, pats, steps, tau, (float*)d_out);
}